// tomatch_22033182228987
// MI455X (gfx1250) — compile-verified
//
#include <hip/hip_runtime.h>
#include <hip/hip_bf16.h>
#include <math.h>

typedef float v2f __attribute__((ext_vector_type(2)));
typedef float v8f __attribute__((ext_vector_type(8)));

#define P_POLY 128
#define N_PRED 128
#define M_GT   1024

// One block per polygon. 256 threads = 8 waves (wave32).
// Wave w owns pred rows [w*16, w*16+16) and loops over 64 gt tiles x 3 levels.
//
// Full squared distance folded into ONE f32 16x16x4 WMMA using the spare K slots:
//   A[r][0..3] = (-2*px_r, -2*py_r, 1, |p_r|^2)
//   B[0..3][c] = ( gx_c,    gy_c,  |g_c|^2, 1 )
//   => D[r][c] = |p_r - g_c|^2   (C = 0 inline)
//
// A layout (32-bit 16x4): lanes 0-15 hold K=0,1 ; lanes 16-31 hold K=2,3 (same row).
// B layout (4x16):        lanes 0-15 hold K=0,1 rows; lanes 16-31 hold K=2,3 rows.
//
// B operand is stored INTERLEAVED in LDS so each tile is one ds_load_b64 straight
// into an even-aligned VGPR pair:
//   bank 0 [0,1024):    (gx, gy)      pairs  -> lanes 0-15
//   bank 1 [1024,2048): (|g|^2, 1.0)  pairs  -> lanes 16-31
__global__ void __launch_bounds__(256)
chamfer_wmma_kernel(const float* __restrict__ pred0,
                    const float* __restrict__ pred1,
                    const float* __restrict__ pred2,
                    const float* __restrict__ gt,
                    float* __restrict__ partial)
{
    const int p    = blockIdx.x;
    const int tid  = threadIdx.x;
    const int wave = tid >> 5;
    const int lane = tid & 31;
    const int col  = lane & 15;
    const bool lo  = (lane < 16);

    const float* preds[3] = {pred0, pred1, pred2};
    const float  wts[3]   = {0.2f, 0.3f, 0.5f};

    __shared__ float2 s_g2[2 * M_GT];   // 16 KB
    __shared__ float  s_part[8];

    const float2* gbase = (const float2*)(gt + (size_t)p * M_GT * 2);
    for (int m = tid; m < M_GT; m += 256) {
        float2 g = gbase[m];
        s_g2[m]        = g;                                          // (gx, gy)
        s_g2[M_GT + m] = make_float2(g.x * g.x + g.y * g.y, 1.0f);   // (gn, 1)
    }
    __syncthreads();

    // Per-lane base index into the interleaved banks (stride 16 pairs per tile).
    const int boff = col + (lo ? 0 : M_GT);

    float wave_acc = 0.0f;

    for (int lev = 0; lev < 3; ++lev) {
        // All 32 lanes load the same pred row as their 16-lane counterpart
        // (coords [1],[2] of 3) -> no divergence, hi lanes build the (1, pnorm) slots.
        const float* pb = preds[lev] + ((size_t)p * N_PRED + wave * 16) * 3;
        const float px = pb[col * 3 + 1];
        const float py = pb[col * 3 + 2];

        v2f a;
        a.x = lo ? (-2.0f * px) : 1.0f;
        a.y = lo ? (-2.0f * py) : (px * px + py * py);

        v8f c = {};  // lowers to inline 0 C operand

        float rmin[8];
#pragma unroll
        for (int r = 0; r < 8; ++r) rmin[r] = 3.0e38f;

        for (int mt = 0; mt < M_GT / 16; ++mt) {
            float2 bb = s_g2[boff + mt * 16];   // one ds_load_b64 per tile
            v2f b;
            b.x = bb.x;
            b.y = bb.y;

            v8f d = __builtin_amdgcn_wmma_f32_16x16x4_f32(
                false, a, false, b, (short)0, c, false, false);

#pragma unroll
            for (int r = 0; r < 8; ++r) rmin[r] = fminf(rmin[r], d[r]);
        }

        // Min over columns: reduce across the 16 lanes of each half
        // (xor offsets 1,2,4,8 never cross the lane-16 boundary).
        // Half 0 lanes then hold rows 0-7 (VGPR r = row r), half 1 rows 8-15.
        float lsum = 0.0f;
#pragma unroll
        for (int r = 0; r < 8; ++r) {
            float v = rmin[r];
            v = fminf(v, __shfl_xor(v, 1, 32));
            v = fminf(v, __shfl_xor(v, 2, 32));
            v = fminf(v, __shfl_xor(v, 4, 32));
            v = fminf(v, __shfl_xor(v, 8, 32));
            lsum += sqrtf(fmaxf(v, 0.0f));
        }

        // lane 0 has rows 0-7 sum, lane 16 rows 8-15 (uniform within each half).
        float other = __shfl_down(lsum, 16, 32);
        if (lane == 0) {
            wave_acc += (lsum + other) * (wts[lev] / (3.0f * (float)N_PRED * (float)P_POLY));
        }
    }

    if (lane == 0) s_part[wave] = wave_acc;
    __syncthreads();
    if (tid == 0) {
        float s = 0.0f;
#pragma unroll
        for (int w = 0; w < 8; ++w) s += s_part[w];
        partial[p] = s;
    }
}

// Deterministic final reduction of 128 per-polygon partials.
__global__ void chamfer_reduce_kernel(const float* __restrict__ partial,
                                      float* __restrict__ out)
{
    if (threadIdx.x == 0) {
        float s = 0.0f;
        for (int i = 0; i < P_POLY; ++i) s += partial[i];
        out[0] = s;
    }
}

extern "C" void kernel_launch(void* const* d_in, const int* in_sizes, int n_in,
                              void* d_out, int out_size, void* d_ws, size_t ws_size,
                              hipStream_t stream) {
    const float* pred0 = (const float*)d_in[0];
    const float* pred1 = (const float*)d_in[1];
    const float* pred2 = (const float*)d_in[2];
    const float* gt    = (const float*)d_in[3];
    float* out = (float*)d_out;
    float* ws  = (float*)d_ws;   // 128 floats of per-polygon partial sums

    chamfer_wmma_kernel<<<P_POLY, 256, 0, stream>>>(pred0, pred1, pred2, gt, ws);
    chamfer_reduce_kernel<<<1, 32, 0, stream>>>(ws, out);
}